// Data_augV4_1838246002702
// MI455X (gfx1250) — compile-verified
//
#include <hip/hip_runtime.h>

// ---------------------------------------------------------------------------
// Data augmentation (256,3,224,224) fp32, per-image transform from sample[n]:
//   0: identity   1: flip W   2: flip H   3: clamp(x*1.5,0,1)
//   4: clamp((x-mean)*1.5+mean,0,1)   5: x<0.5 ? x : 1-x
//
// Memory-bound: ~302 MB total traffic -> ~13us at 23.3 TB/s. Strategy:
//   * pass 1: per-image mean ONLY for sample==4 images, reduced with
//     V_WMMA_F32_16X16X4_F32 (B = ones -> C accumulates row sums; 64 f32
//     summed per wmma per wave). One b128 load feeds two WMMAs.
//   * pass 2: one float4 per thread, coalesced b128 gather for flips,
//     non-temporal b128 stores (output is write-once; keep L2 for input).
// ---------------------------------------------------------------------------

typedef float v2f __attribute__((ext_vector_type(2)));
typedef float v4f __attribute__((ext_vector_type(4)));
typedef float v8f __attribute__((ext_vector_type(8)));

#define IMG_ELEMS  (3 * 224 * 224)          // 150528 floats per image
#define IMG_VEC4   (IMG_ELEMS / 4)          // 37632 float4 per image
#define MEAN_ITERS (IMG_ELEMS / 1024)       // 147: 8 waves * 128 elems / iter
#define TF_BLOCKS  (IMG_VEC4 / 256)         // 147 blocks of 256 threads

// ---------------------------------------------------------------------------
// Pass 1: per-image mean via WMMA row-sum accumulation.
// D[m][n] = sum_k A[m][k]*1 + C[m][n]  ->  C accumulates row sums of A.
// Every element of the wave's 128-float tile lands in exactly one A slot of
// one of the two WMMAs, and every column of C replicates the row sum, so
// sum(all C) = 16 * true sum.
// ---------------------------------------------------------------------------
__global__ __launch_bounds__(256) void mean_kernel(const float* __restrict__ x,
                                                   const int* __restrict__ sample,
                                                   float* __restrict__ mean_out) {
    const int img = blockIdx.x;
    const int tid = threadIdx.x;

    const int tf = sample[img];          // uniform per block
    if (tf != 4) {                       // mean only needed for contrast
        if (tid == 0) mean_out[img] = 0.0f;
        return;
    }

    const int wave = tid >> 5;           // wave32
    const int lane = tid & 31;
    const float* base = x + (size_t)img * IMG_ELEMS + (size_t)(wave * 128 + lane * 4);

    const v2f ones2 = {1.0f, 1.0f};
    v8f acc = {};                        // C/D accumulator (8 VGPRs)

#pragma unroll 2
    for (int i = 0; i < MEAN_ITERS; ++i) {
        v4f v = *(const v4f*)(base + i * 1024);  // global_load_b128, coalesced
        v2f a0 = {v[0], v[1]};
        v2f a1 = {v[2], v[3]};
        // 8 args: (neg_a, A, neg_b, B, c_mod, C, reuse_a, reuse_b)
        acc = __builtin_amdgcn_wmma_f32_16x16x4_f32(
            false, a0, false, ones2, (short)0, acc, false, false);
        acc = __builtin_amdgcn_wmma_f32_16x16x4_f32(
            false, a1, false, ones2, (short)0, acc, false, false);
    }

    float partial = acc[0] + acc[1] + acc[2] + acc[3] +
                    acc[4] + acc[5] + acc[6] + acc[7];

    __shared__ float red[256];
    red[tid] = partial;
    __syncthreads();
#pragma unroll
    for (int s = 128; s > 0; s >>= 1) {
        if (tid < s) red[tid] += red[tid + s];
        __syncthreads();
    }
    if (tid == 0) {
        // each element counted once per replicated column (16x)
        mean_out[img] = red[0] * (1.0f / (16.0f * (float)IMG_ELEMS));
    }
}

// ---------------------------------------------------------------------------
// Pass 2: streaming transform, one float4 per thread.
// ---------------------------------------------------------------------------
__device__ __forceinline__ float clamp01(float v) {
    return fminf(fmaxf(v, 0.0f), 1.0f);
}

__global__ __launch_bounds__(256) void tf_kernel(const float* __restrict__ x,
                                                 const int* __restrict__ sample,
                                                 const float* __restrict__ means,
                                                 float* __restrict__ out) {
    const int img = blockIdx.y;
    const int tf  = sample[img];                       // uniform per block
    const int g   = blockIdx.x * 256 + threadIdx.x;    // float4 index in image

    const size_t imgOff = (size_t)img * IMG_ELEMS;
    const v4f* in4 = (const v4f*)(x + imgOff);
    v4f*      out4 = (v4f*)(out + imgOff);

    const int w4   = g % 56;        // float4 column (W/4 = 56)
    const int rest = g / 56;        // c*224 + h  in [0, 672)

    v4f r;
    switch (tf) {
    case 1: {   // flip along W: mirrored float4, reversed components
        v4f v = in4[rest * 56 + (55 - w4)];
        r[0] = v[3]; r[1] = v[2]; r[2] = v[1]; r[3] = v[0];
        break;
    }
    case 2: {   // flip along H: mirrored row, same column
        const int h = rest % 224;
        const int c = rest / 224;
        r = in4[(c * 224 + (223 - h)) * 56 + w4];
        break;
    }
    case 3: {   // brightness
        v4f v = in4[g];
        r[0] = clamp01(v[0] * 1.5f); r[1] = clamp01(v[1] * 1.5f);
        r[2] = clamp01(v[2] * 1.5f); r[3] = clamp01(v[3] * 1.5f);
        break;
    }
    case 4: {   // contrast about per-image mean
        const float m = means[img];
        v4f v = in4[g];
        r[0] = clamp01((v[0] - m) * 1.5f + m);
        r[1] = clamp01((v[1] - m) * 1.5f + m);
        r[2] = clamp01((v[2] - m) * 1.5f + m);
        r[3] = clamp01((v[3] - m) * 1.5f + m);
        break;
    }
    case 5: {   // solarize
        v4f v = in4[g];
        r[0] = (v[0] < 0.5f) ? v[0] : 1.0f - v[0];
        r[1] = (v[1] < 0.5f) ? v[1] : 1.0f - v[1];
        r[2] = (v[2] < 0.5f) ? v[2] : 1.0f - v[2];
        r[3] = (v[3] < 0.5f) ? v[3] : 1.0f - v[3];
        break;
    }
    default:    // identity
        r = in4[g];
        break;
    }

    // write-once output: non-temporal b128 store, keep L2 for the input side
    __builtin_nontemporal_store(r, &out4[g]);
}

// ---------------------------------------------------------------------------
// Host entry
// ---------------------------------------------------------------------------
extern "C" void kernel_launch(void* const* d_in, const int* in_sizes, int n_in,
                              void* d_out, int out_size, void* d_ws, size_t ws_size,
                              hipStream_t stream) {
    const float* x      = (const float*)d_in[0];   // (256,3,224,224) f32
    const int*   sample = (const int*)d_in[1];     // (256,) i32 in [0,6)
    float*       out    = (float*)d_out;
    float*       means  = (float*)d_ws;            // 256 floats of scratch

    const int n_img = in_sizes[1];                 // 256

    mean_kernel<<<n_img, 256, 0, stream>>>(x, sample, means);

    dim3 grid(TF_BLOCKS, n_img);                   // 147 x 256
    tf_kernel<<<grid, 256, 0, stream>>>(x, sample, means, out);
}